// TrivialisedDiffusion_39307540693614
// MI455X (gfx1250) — compile-verified
//
#include <hip/hip_runtime.h>
#include <cstdint>

// ---------------------------------------------------------------------------
// TrivialisedDiffusion on MI455X (gfx1250).
// Memory-bound streaming + 3 chained segment-mean reductions (index sorted).
// 5 kernels on one stream:
//   K0: zero 13*S-float segment tables in d_ws
//   K1: seg-sums of eps_v, eps_r, counts        (LDS ds_add_f32 pre-reduce)
//   K2: v_t, mu_r, r_t' ; seg-sum of r_t'       (writes ch1, ch3(tmp mu_r), ch0(tmp r_t'))
//   K3: r_t, wn ; seg-sum of wn                 (writes ch2, ch0(tmp wn))
//   K4: f_t, target                             (final ch0, ch3; NT stores)
// CDNA5-specific paths: global_load_async_to_lds_b32 + s_wait_asynccnt for
// index staging, ds_add_f32 LDS atomics, NT temporal hints on dead streams.
// ---------------------------------------------------------------------------

#define BLK   256
#define NSLOT 32
#define EPSV  1e-5f

// ---- async global->LDS index staging (ASYNCcnt path) ----------------------
__device__ __forceinline__ void async_i32_to_lds(const int* g, int* l) {
#if __has_builtin(__builtin_amdgcn_global_load_async_to_lds_b32)
  __builtin_amdgcn_global_load_async_to_lds_b32(const_cast<int*>(g), l, 0, 0);
#else
  typedef __attribute__((address_space(3))) int lint_t;
  unsigned lds_off = (unsigned)(__SIZE_TYPE__)(lint_t*)l;
  asm volatile("global_load_async_to_lds_b32 %0, %1, off"
               :: "v"(lds_off), "v"((const void*)g) : "memory");
#endif
}

__device__ __forceinline__ void wait_async_lds() {
#if __has_builtin(__builtin_amdgcn_s_wait_asynccnt)
  __builtin_amdgcn_s_wait_asynccnt(0);
#else
  asm volatile("s_wait_asynccnt 0x0" ::: "memory");
#endif
}

// ---- math helpers ---------------------------------------------------------
__device__ __forceinline__ float wsu(float x) {       // wrap_signed_unit
  float y = x + 0.5f;
  return y - floorf(y) - 0.5f;
}
__device__ __forceinline__ float wfrac(float x) {     // wrap_fractional
  return x - floorf(x);
}

// softmax-weighted wrapped-normal score, ks = -3..3
__device__ __forceinline__ float d_log_wrapped_normal(float r, float mu, float inv_var) {
  const float b = r - mu;
  float d[7], l[7];
  float m = -3.4e38f;
#pragma unroll
  for (int k = 0; k < 7; ++k) {
    d[k] = b + (float)(k - 3);
    l[k] = -0.5f * d[k] * d[k] * inv_var;
    m = fmaxf(m, l[k]);
  }
  float den = 0.f, num = 0.f;
#pragma unroll
  for (int k = 0; k < 7; ++k) {
    float w = __expf(l[k] - m);
    den += w;
    num -= d[k] * inv_var * w;
  }
  return num / den;
}

// workspace layout: [sum_ev 3S][sum_er 3S][cnt S][sum_rtp 3S][sum_wn 3S] = 13S
// ---------------------------------------------------------------------------
__global__ void k0_zero(float* __restrict__ ws, const int* __restrict__ pS) {
  long tot = 13L * (long)(*pS);
  for (long x = (long)blockIdx.x * blockDim.x + threadIdx.x; x < tot;
       x += (long)gridDim.x * blockDim.x)
    ws[x] = 0.f;
}

// ---------------------------------------------------------------------------
__global__ void k1_sums(const float* __restrict__ ev, const float* __restrict__ er,
                        const int* __restrict__ idx, const int* __restrict__ pS,
                        int N, float* __restrict__ ws) {
  __shared__ int   s_idx[BLK];
  __shared__ float slot[NSLOT * 7];          // ev.xyz, er.xyz, cnt
  const int tid = threadIdx.x;
  const int i   = blockIdx.x * BLK + tid;
  async_i32_to_lds(idx + (i < N ? i : N - 1), &s_idx[tid]);
  for (int k = tid; k < NSLOT * 7; k += BLK) slot[k] = 0.f;
  wait_async_lds();
  __syncthreads();

  const int S = *pS;
  float* sev = ws;
  float* ser = ws + 3 * (size_t)S;
  float* cnt = ws + 6 * (size_t)S;
  const int base = s_idx[0];

  if (i < N) {
    const int seg = s_idx[tid];
    const size_t i3 = (size_t)i * 3;
    float e0 = ev[i3], e1 = ev[i3 + 1], e2 = ev[i3 + 2];
    float r0 = er[i3], r1 = er[i3 + 1], r2 = er[i3 + 2];
    const int sl = seg - base;
    if ((unsigned)sl < (unsigned)NSLOT) {        // sorted index: common path
      float* p = &slot[sl * 7];
      atomicAdd(p + 0, e0); atomicAdd(p + 1, e1); atomicAdd(p + 2, e2);
      atomicAdd(p + 3, r0); atomicAdd(p + 4, r1); atomicAdd(p + 5, r2);
      atomicAdd(p + 6, 1.0f);
    } else {                                     // robustness fallback
      atomicAdd(&sev[3 * (size_t)seg + 0], e0);
      atomicAdd(&sev[3 * (size_t)seg + 1], e1);
      atomicAdd(&sev[3 * (size_t)seg + 2], e2);
      atomicAdd(&ser[3 * (size_t)seg + 0], r0);
      atomicAdd(&ser[3 * (size_t)seg + 1], r1);
      atomicAdd(&ser[3 * (size_t)seg + 2], r2);
      atomicAdd(&cnt[seg], 1.0f);
    }
  }
  __syncthreads();
  if (tid < NSLOT) {
    const float c = slot[tid * 7 + 6];
    if (c != 0.f) {
      const int seg = base + tid;
      atomicAdd(&sev[3 * (size_t)seg + 0], slot[tid * 7 + 0]);
      atomicAdd(&sev[3 * (size_t)seg + 1], slot[tid * 7 + 1]);
      atomicAdd(&sev[3 * (size_t)seg + 2], slot[tid * 7 + 2]);
      atomicAdd(&ser[3 * (size_t)seg + 0], slot[tid * 7 + 3]);
      atomicAdd(&ser[3 * (size_t)seg + 1], slot[tid * 7 + 4]);
      atomicAdd(&ser[3 * (size_t)seg + 2], slot[tid * 7 + 5]);
      atomicAdd(&cnt[seg], c);
    }
  }
}

// ---------------------------------------------------------------------------
__global__ void k2_vt(const float* __restrict__ t,  const float* __restrict__ v0,
                      const float* __restrict__ ev, const float* __restrict__ er,
                      const int* __restrict__ idx,  const int* __restrict__ pS,
                      int N, float* __restrict__ ws, float* __restrict__ out) {
  __shared__ int   s_idx[BLK];
  __shared__ float slot[NSLOT * 3];
  __shared__ int   touched[NSLOT];
  const int tid = threadIdx.x;
  const int i   = blockIdx.x * BLK + tid;
  async_i32_to_lds(idx + (i < N ? i : N - 1), &s_idx[tid]);
  for (int k = tid; k < NSLOT * 3; k += BLK) slot[k] = 0.f;
  if (tid < NSLOT) touched[tid] = 0;
  wait_async_lds();
  __syncthreads();

  const int S = *pS;
  const float* sev = ws;
  const float* ser = ws + 3 * (size_t)S;
  const float* cnt = ws + 6 * (size_t)S;
  float*       srt = ws + 7 * (size_t)S;
  const size_t CH = (size_t)N * 3;
  float* ch0 = out;            // tmp: r_t'
  float* ch1 = out + CH;       // final: v_t
  float* ch3 = out + 3 * CH;   // tmp: mu_r
  const int base = s_idx[0];

  float rtp[3] = {0.f, 0.f, 0.f};
  int   seg = 0;
  if (i < N) {
    seg = s_idx[tid];
    const size_t i3 = (size_t)i * 3;
    const float tt = t[i];
    const float ts = 2.0f * tt;
    const float e_t = __expf(-ts);
    const float sigma_v = sqrtf(fmaxf(1.0f - e_t * e_t, EPSV));
    const float coeff = (1.0f - e_t) / (1.0f + e_t);
    const float sigma_r = sqrtf(fmaxf(2.0f * ts + 8.0f * e_t / (1.0f + e_t) - 4.0f, EPSV));
    const float invc = 1.0f / fmaxf(cnt[seg], 1.0f);
#pragma unroll
    for (int c = 0; c < 3; ++c) {
      const float v0c = __builtin_nontemporal_load(&v0[i3 + c]);  // single-use stream
      const float evc = ev[i3 + c] - sev[3 * (size_t)seg + c] * invc;
      const float erc = er[i3 + c] - ser[3 * (size_t)seg + c] * invc;
      const float vt  = e_t * v0c + sigma_v * evc;
      const float mu  = coeff * (vt + v0c);
      const float rp  = wsu(mu + sigma_r * erc);
      ch1[i3 + c] = vt;
      ch3[i3 + c] = mu;
      ch0[i3 + c] = rp;
      rtp[c] = rp;
    }
    const int sl = seg - base;
    if ((unsigned)sl < (unsigned)NSLOT) {
      atomicAdd(&slot[sl * 3 + 0], rtp[0]);
      atomicAdd(&slot[sl * 3 + 1], rtp[1]);
      atomicAdd(&slot[sl * 3 + 2], rtp[2]);
      touched[sl] = 1;
    } else {
      atomicAdd(&srt[3 * (size_t)seg + 0], rtp[0]);
      atomicAdd(&srt[3 * (size_t)seg + 1], rtp[1]);
      atomicAdd(&srt[3 * (size_t)seg + 2], rtp[2]);
    }
  }
  __syncthreads();
  if (tid < NSLOT && touched[tid]) {
    const int sg = base + tid;
    atomicAdd(&srt[3 * (size_t)sg + 0], slot[tid * 3 + 0]);
    atomicAdd(&srt[3 * (size_t)sg + 1], slot[tid * 3 + 1]);
    atomicAdd(&srt[3 * (size_t)sg + 2], slot[tid * 3 + 2]);
  }
}

// ---------------------------------------------------------------------------
__global__ void k3_rt(const float* __restrict__ t, const int* __restrict__ idx,
                      const int* __restrict__ pS, int N,
                      float* __restrict__ ws, float* __restrict__ out) {
  __shared__ int   s_idx[BLK];
  __shared__ float slot[NSLOT * 3];
  __shared__ int   touched[NSLOT];
  const int tid = threadIdx.x;
  const int i   = blockIdx.x * BLK + tid;
  async_i32_to_lds(idx + (i < N ? i : N - 1), &s_idx[tid]);
  for (int k = tid; k < NSLOT * 3; k += BLK) slot[k] = 0.f;
  if (tid < NSLOT) touched[tid] = 0;
  wait_async_lds();
  __syncthreads();

  const int S = *pS;
  const float* cnt = ws + 6 * (size_t)S;
  const float* srt = ws + 7 * (size_t)S;
  float*       swn = ws + 10 * (size_t)S;
  const size_t CH = (size_t)N * 3;
  float* ch0 = out;            // in: r_t'  -> out: wn (tmp)
  float* ch2 = out + 2 * CH;   // final: r_t
  float* ch3 = out + 3 * CH;   // in: mu_r
  const int base = s_idx[0];

  float wn[3] = {0.f, 0.f, 0.f};
  int   seg = 0;
  if (i < N) {
    seg = s_idx[tid];
    const size_t i3 = (size_t)i * 3;
    const float tt = t[i];
    const float ts = 2.0f * tt;
    const float e_t = __expf(-ts);
    const float coeff = (1.0f - e_t) / (1.0f + e_t);
    const float sigma_r = sqrtf(fmaxf(2.0f * ts + 8.0f * e_t / (1.0f + e_t) - 4.0f, EPSV));
    const float inv_var = 1.0f / (sigma_r * sigma_r);
    const float invc = 1.0f / fmaxf(cnt[seg], 1.0f);
#pragma unroll
    for (int c = 0; c < 3; ++c) {
      const float rp = ch0[i3 + c];
      const float mu = ch3[i3 + c];
      const float rt = wsu(rp - srt[3 * (size_t)seg + c] * invc);
      ch2[i3 + c] = rt;
      const float w = coeff * d_log_wrapped_normal(rt, mu, inv_var);
      ch0[i3 + c] = w;   // same thread already consumed rp
      wn[c] = w;
    }
    const int sl = seg - base;
    if ((unsigned)sl < (unsigned)NSLOT) {
      atomicAdd(&slot[sl * 3 + 0], wn[0]);
      atomicAdd(&slot[sl * 3 + 1], wn[1]);
      atomicAdd(&slot[sl * 3 + 2], wn[2]);
      touched[sl] = 1;
    } else {
      atomicAdd(&swn[3 * (size_t)seg + 0], wn[0]);
      atomicAdd(&swn[3 * (size_t)seg + 1], wn[1]);
      atomicAdd(&swn[3 * (size_t)seg + 2], wn[2]);
    }
  }
  __syncthreads();
  if (tid < NSLOT && touched[tid]) {
    const int sg = base + tid;
    atomicAdd(&swn[3 * (size_t)sg + 0], slot[tid * 3 + 0]);
    atomicAdd(&swn[3 * (size_t)sg + 1], slot[tid * 3 + 1]);
    atomicAdd(&swn[3 * (size_t)sg + 2], slot[tid * 3 + 2]);
  }
}

// ---------------------------------------------------------------------------
__global__ void k4_final(const float* __restrict__ t, const float* __restrict__ f0,
                         const int* __restrict__ idx, const int* __restrict__ pS,
                         int N, const float* __restrict__ ws, float* __restrict__ out) {
  __shared__ int s_idx[BLK];
  const int tid = threadIdx.x;
  const int i   = blockIdx.x * BLK + tid;
  async_i32_to_lds(idx + (i < N ? i : N - 1), &s_idx[tid]);
  wait_async_lds();
  __syncthreads();

  const int S = *pS;
  const float* cnt = ws + 6 * (size_t)S;
  const float* swn = ws + 10 * (size_t)S;
  const size_t CH = (size_t)N * 3;
  float* ch0 = out;            // in: wn   -> out: f_t (final)
  float* ch1 = out + CH;       // in: v_t
  float* ch2 = out + 2 * CH;   // in: r_t
  float* ch3 = out + 3 * CH;   // out: target (final)

  if (i < N) {
    const int seg = s_idx[tid];
    const size_t i3 = (size_t)i * 3;
    const float tt = t[i];
    const float ts = 2.0f * tt;
    const float e_t = __expf(-ts);
    const float sigma_v = sqrtf(fmaxf(1.0f - e_t * e_t, EPSV));
    const float sv = fmaxf(sigma_v, EPSV);
    const float inv_sv2 = 1.0f / (sv * sv);
    const float invc = 1.0f / fmaxf(cnt[seg], 1.0f);
#pragma unroll
    for (int c = 0; c < 3; ++c) {
      const float wnc = ch0[i3 + c] - swn[3 * (size_t)seg + c] * invc;
      const float vt  = ch1[i3 + c];
      const float tgt = -vt * inv_sv2 + wnc;
      __builtin_nontemporal_store(tgt, &ch3[i3 + c]);        // never re-read
      const float f0c = __builtin_nontemporal_load(&f0[i3 + c]);
      const float ft  = wfrac(f0c + ch2[i3 + c]);
      __builtin_nontemporal_store(ft, &ch0[i3 + c]);         // never re-read
    }
  }
}

// ---------------------------------------------------------------------------
extern "C" void kernel_launch(void* const* d_in, const int* in_sizes, int n_in,
                              void* d_out, int out_size, void* d_ws, size_t ws_size,
                              hipStream_t stream) {
  const float* t  = (const float*)d_in[0];
  const float* f0 = (const float*)d_in[1];
  const float* v0 = (const float*)d_in[2];
  const float* ev = (const float*)d_in[3];
  const float* er = (const float*)d_in[4];
  const int*   idx = (const int*)d_in[5];
  const int*   pS  = (const int*)d_in[6];
  float* ws  = (float*)d_ws;
  float* out = (float*)d_out;

  const int N  = in_sizes[0];
  const int nb = (N + BLK - 1) / BLK;

  k0_zero<<<416, BLK, 0, stream>>>(ws, pS);
  k1_sums<<<nb, BLK, 0, stream>>>(ev, er, idx, pS, N, ws);
  k2_vt  <<<nb, BLK, 0, stream>>>(t, v0, ev, er, idx, pS, N, ws, out);
  k3_rt  <<<nb, BLK, 0, stream>>>(t, idx, pS, N, ws, out);
  k4_final<<<nb, BLK, 0, stream>>>(t, f0, idx, pS, N, ws, out);
}